// GNNClassifier_81046032876152
// MI455X (gfx1250) — compile-verified
//
#include <hip/hip_runtime.h>

typedef float v2f __attribute__((ext_vector_type(2)));
typedef float v8f __attribute__((ext_vector_type(8)));

#define IN_DIM  128
#define HID_DIM 32
#define OUT_DIM 8

// ---------------------------------------------------------------------------
// 1) init: deg=1 (self loop), h accumulator = 0, out seeded with bias b2
// ---------------------------------------------------------------------------
__global__ void __launch_bounds__(256) init_kernel(float* __restrict__ deg,
                                                   float* __restrict__ h,
                                                   float* __restrict__ out,
                                                   const float* __restrict__ b2,
                                                   int n) {
    int i = blockIdx.x * blockDim.x + threadIdx.x;   // grid covers n*HID_DIM
    if (i < n) deg[i] = 1.0f;                        // self-loop degree contribution
    if (i < n * OUT_DIM) out[i] = b2[i & (OUT_DIM - 1)];
    if (i < n * HID_DIM) h[i] = 0.0f;
}

// ---------------------------------------------------------------------------
// 2) degree accumulation over destination nodes
// ---------------------------------------------------------------------------
__global__ void __launch_bounds__(256) deg_kernel(const int* __restrict__ dstIdx,
                                                  float* __restrict__ deg, int E) {
    int e = blockIdx.x * blockDim.x + threadIdx.x;
    if (e < E) atomicAdd(&deg[dstIdx[e]], 1.0f);
}

// 3) deg -> deg^{-1/2} in place
__global__ void __launch_bounds__(256) rsqrt_kernel(float* __restrict__ deg, int n) {
    int i = blockIdx.x * blockDim.x + threadIdx.x;
    if (i < n) deg[i] = rsqrtf(deg[i]);
}

// ---------------------------------------------------------------------------
// 4) hw = x @ W1   [n,128] x [128,32]  via V_WMMA_F32_16X16X4_F32
//    One wave per 16-row tile; two 16x16 accumulators (cols 0-15, 16-31).
//    A-frag (16x4 f32): lanes 0-15 -> K=k,k+1 ; lanes 16-31 -> K=k+2,k+3.
//    B-frag (4x16 f32): same K split, N = lane&15.
//    D (16x16 f32): VGPR i -> row i (+8 for lanes>=16), col = lane&15.
// ---------------------------------------------------------------------------
__global__ void __launch_bounds__(256) gemm1_kernel(const float* __restrict__ x,
                                                    const float* __restrict__ W1,
                                                    float* __restrict__ hw, int n) {
    const int lane = threadIdx.x & 31;
    const int half = lane >> 4;          // 0: K pair (0,1)   1: K pair (2,3)
    const int l16  = lane & 15;
    const int tile = blockIdx.x * 8 + (threadIdx.x >> 5);
    const int ntiles = (n + 15) >> 4;
    if (tile >= ntiles) return;          // wave-uniform: EXEC stays all-ones

    int row = tile * 16 + l16;
    if (row >= n) row = n - 1;           // tail-safe load clamp (stores guarded)
    const float* xr = x + row * IN_DIM;

    v8f c0 = {}; v8f c1 = {};
    #pragma unroll 8
    for (int k = 0; k < IN_DIM; k += 4) {
        const int kk = k + 2 * half;
        v2f a;  a.x  = xr[kk];                       a.y  = xr[kk + 1];
        v2f b0; b0.x = W1[kk * HID_DIM + l16];       b0.y = W1[(kk + 1) * HID_DIM + l16];
        v2f b1; b1.x = W1[kk * HID_DIM + 16 + l16];  b1.y = W1[(kk + 1) * HID_DIM + 16 + l16];
        c0 = __builtin_amdgcn_wmma_f32_16x16x4_f32(false, a, false, b0, (short)0, c0, false, false);
        c1 = __builtin_amdgcn_wmma_f32_16x16x4_f32(false, a, false, b1, (short)0, c1, false, false);
    }
    #pragma unroll
    for (int i = 0; i < 8; ++i) {
        int orow = tile * 16 + i + 8 * half;
        if (orow < n) {
            hw[orow * HID_DIM + l16]      = c0[i];
            hw[orow * HID_DIM + 16 + l16] = c1[i];
        }
    }
}

// ---------------------------------------------------------------------------
// 5) layer-1 aggregation: one wave per edge, lane = hidden feature (32-wide)
//    edges [0,E) are real edges, [E, E+n) are self-loops.
// ---------------------------------------------------------------------------
__global__ void __launch_bounds__(256) agg1_kernel(const int* __restrict__ srcIdx,
                                                   const int* __restrict__ dstIdx,
                                                   const float* __restrict__ dinv,
                                                   const float* __restrict__ hw,
                                                   float* __restrict__ h,
                                                   int E, int n) {
    int tid  = blockIdx.x * blockDim.x + threadIdx.x;
    int e    = tid >> 5;
    int lane = tid & 31;
    if (e >= E + n) return;
    int s, d;
    if (e < E) { s = srcIdx[e]; d = dstIdx[e]; }
    else       { s = e - E;     d = s; }
    float nrm = dinv[s] * dinv[d];
    float val = hw[s * HID_DIM + lane] * nrm;       // coalesced 128B gather (L2-resident)
    atomicAdd(&h[d * HID_DIM + lane], val);         // coalesced 128B scatter-add
}

// 6) h = relu(h + b1)
__global__ void __launch_bounds__(256) bias_relu_kernel(float* __restrict__ h,
                                                        const float* __restrict__ b1, int n) {
    int i = blockIdx.x * blockDim.x + threadIdx.x;
    if (i < n * HID_DIM) h[i] = fmaxf(h[i] + b1[i & (HID_DIM - 1)], 0.0f);
}

// ---------------------------------------------------------------------------
// 7) g = h @ W2   [n,32] x [32,8]; N padded to 16 with zero columns.
// ---------------------------------------------------------------------------
__global__ void __launch_bounds__(256) gemm2_kernel(const float* __restrict__ h,
                                                    const float* __restrict__ W2,
                                                    float* __restrict__ g, int n) {
    const int lane = threadIdx.x & 31;
    const int half = lane >> 4;
    const int l16  = lane & 15;
    const int tile = blockIdx.x * 8 + (threadIdx.x >> 5);
    const int ntiles = (n + 15) >> 4;
    if (tile >= ntiles) return;

    int row = tile * 16 + l16;
    if (row >= n) row = n - 1;
    const float* hr = h + row * HID_DIM;
    const bool live = (l16 < OUT_DIM);               // selects, not branches -> EXEC intact

    v8f c = {};
    #pragma unroll
    for (int k = 0; k < HID_DIM; k += 4) {
        const int kk = k + 2 * half;
        v2f a; a.x = hr[kk]; a.y = hr[kk + 1];
        v2f b;
        b.x = live ? W2[kk * OUT_DIM + l16]       : 0.0f;
        b.y = live ? W2[(kk + 1) * OUT_DIM + l16] : 0.0f;
        c = __builtin_amdgcn_wmma_f32_16x16x4_f32(false, a, false, b, (short)0, c, false, false);
    }
    if (live) {
        #pragma unroll
        for (int i = 0; i < 8; ++i) {
            int orow = tile * 16 + i + 8 * half;
            if (orow < n) g[orow * OUT_DIM + l16] = c[i];
        }
    }
}

// ---------------------------------------------------------------------------
// 8) layer-2 aggregation: 8 lanes per edge (4 edges per wave32), into d_out
//    (d_out already holds b2 broadcast from init_kernel)
// ---------------------------------------------------------------------------
__global__ void __launch_bounds__(256) agg2_kernel(const int* __restrict__ srcIdx,
                                                   const int* __restrict__ dstIdx,
                                                   const float* __restrict__ dinv,
                                                   const float* __restrict__ g,
                                                   float* __restrict__ out,
                                                   int E, int n) {
    int tid = blockIdx.x * blockDim.x + threadIdx.x;
    int e = tid >> 3;
    int f = tid & 7;
    if (e >= E + n) return;
    int s, d;
    if (e < E) { s = srcIdx[e]; d = dstIdx[e]; }
    else       { s = e - E;     d = s; }
    float nrm = dinv[s] * dinv[d];
    atomicAdd(&out[d * OUT_DIM + f], g[s * OUT_DIM + f] * nrm);
}

// ---------------------------------------------------------------------------
extern "C" void kernel_launch(void* const* d_in, const int* in_sizes, int n_in,
                              void* d_out, int out_size, void* d_ws, size_t ws_size,
                              hipStream_t stream) {
    const float* x  = (const float*)d_in[0];
    const int*   ei = (const int*)d_in[1];      // [2,E]: row0 = src, row1 = dst
    const float* W1 = (const float*)d_in[2];
    const float* b1 = (const float*)d_in[3];
    const float* W2 = (const float*)d_in[4];
    const float* b2 = (const float*)d_in[5];
    float* out = (float*)d_out;

    const int n = in_sizes[0] / IN_DIM;
    const int E = in_sizes[1] / 2;
    const int* srcIdx = ei;
    const int* dstIdx = ei + E;

    // workspace layout (all L2-resident: ~29.2 MB total for N=100k)
    float* deg = (float*)d_ws;                        // n floats -> becomes dinv
    float* hw  = deg + n;                             // n * 32
    float* h   = hw + (size_t)n * HID_DIM;            // n * 32
    float* g   = h  + (size_t)n * HID_DIM;            // n * 8

    const int ET = E + n;
    const dim3 blk(256);

    init_kernel<<<dim3((n * HID_DIM + 255) / 256), blk, 0, stream>>>(deg, h, out, b2, n);
    deg_kernel<<<dim3((E + 255) / 256), blk, 0, stream>>>(dstIdx, deg, E);
    rsqrt_kernel<<<dim3((n + 255) / 256), blk, 0, stream>>>(deg, n);

    const int ntiles = (n + 15) / 16;
    gemm1_kernel<<<dim3((ntiles + 7) / 8), blk, 0, stream>>>(x, W1, hw, n);

    long t1 = (long)ET * 32;
    agg1_kernel<<<dim3((unsigned)((t1 + 255) / 256)), blk, 0, stream>>>(srcIdx, dstIdx, deg, hw, h, E, n);
    bias_relu_kernel<<<dim3((n * HID_DIM + 255) / 256), blk, 0, stream>>>(h, b1, n);

    gemm2_kernel<<<dim3((ntiles + 7) / 8), blk, 0, stream>>>(h, W2, g, n);

    long t2 = (long)ET * 8;
    agg2_kernel<<<dim3((unsigned)((t2 + 255) / 256)), blk, 0, stream>>>(srcIdx, dstIdx, deg, g, out, E, n);
}